// Net_21174188769694
// MI455X (gfx1250) — compile-verified
//
#include <hip/hip_runtime.h>

typedef __attribute__((ext_vector_type(2))) float v2f;
typedef __attribute__((ext_vector_type(8))) float v8f;

#define HH 496
#define HW 246016           // 496*496
#define NW 31               // windows per side
#define NWIN 961            // 31*31
#define PADW 512            // padded row stride (power of two)
#define PADH 504            // padded rows (covers y+dy up to 500)
#define PIF 3.14159265358979323846f

// ---------------------------------------------------------------------------
// Stage 1: Gabor bank, padded to 80 channels x 28 taps (zeros beyond 72 x 25)
// ---------------------------------------------------------------------------
__global__ void gabor_kernel(const float* __restrict__ sigmas,
                             const float* __restrict__ lambdas,
                             float* __restrict__ Fp) {
    for (int i = threadIdx.x; i < 80 * 28; i += 256) {
        int c = i / 28, k = i % 28;
        float v = 0.0f;
        if (c < 72 && k < 25) {
            int s  = c / 24;          // sigma index
            int l  = (c / 8) % 3;     // lambda index
            int ti = c % 8;           // theta index
            float sig = sigmas[s];
            float lam = lambdas[l];
            float th  = (float)(ti + 1) * (PIF / 8.0f);   // THETAS[ti]
            float yv  = (float)(k / 5) - 2.0f;
            float xv  = (float)(k % 5) - 2.0f;
            float st = sinf(th), ct = cosf(th);
            float yth = -xv * st + yv * ct;
            float xth =  xv * ct + yv * st;
            float sx2 = sig * sig;
            float sy  = sig * 2.0f;            // sigma / GAMMA, GAMMA = 0.5
            float sy2 = sy * sy;
            v = expf(-0.5f * (xth * xth / sx2 + yth * yth / sy2)) *
                cosf(2.0f * PIF * xth / lam + 0.5f * PIF);
        }
        Fp[i] = v;
    }
}

// ---------------------------------------------------------------------------
// Stage 1b: zero-pad image into 504x512 buffer; pad[r][c] = img[r-2][c-2].
// Removes all bounds checks from the WMMA conv's tap loads.
// ---------------------------------------------------------------------------
__global__ void pad_kernel(const float* __restrict__ img,
                           float* __restrict__ pad) {
    int i = blockIdx.x * 256 + threadIdx.x;      // 504*512 = 258048 exact
    int r = i >> 9, c = i & (PADW - 1);
    int y = r - 2, x = c - 2;
    float v = 0.0f;
    if (y >= 0 && y < HH && x >= 0 && x < HH) v = img[y * HH + x];
    pad[i] = v;
}

// ---------------------------------------------------------------------------
// Stage 2: 5x5 conv (72 ch) via V_WMMA_F32_16X16X4_F32 + theta-max -> 9 maps.
// One wave = 16 consecutive pixels in a row, all 80 (padded) channels.
// A 16x4 layout: lane p holds F[m*16+p][k0+2*hi+{0,1}]        (hi = lane>>4)
// B 4x16 layout: vgpr j, half hi holds P[k0+2*hi+j][lane&15]
// D 16x16:       vgpr v, lane l holds ch = m*16 + v + 8*hi, pixel = l&15
//   => one accumulator tile per lane is exactly one theta-group of 8 -> fmax.
// Taps 25..27 load real (finite) pixels but hit zero filter rows.
// ---------------------------------------------------------------------------
__global__ void __launch_bounds__(256)
conv_wmma_kernel(const float* __restrict__ pad, const float* __restrict__ Fp,
                 float* __restrict__ out9) {
    const int lane = threadIdx.x & 31;
    const int wave = (int)((blockIdx.x * 256u + threadIdx.x) >> 5);
    const int y  = wave / NW;
    const int x0 = (wave % NW) << 4;
    const int p  = lane & 15;
    const int hi = lane >> 4;
    const int x  = x0 + p;

    const float* bbase = pad + (size_t)y * PADW + x;   // tap (dy,dx) at +dy*512+dx
    const float* abase = Fp + p * 28 + 2 * hi;         // v2f-aligned (byte off % 8 == 0)

    // B (im2col) loads: unconditional, compile-time offsets selected by hi.
    v2f b[7];
#pragma unroll
    for (int kc = 0; kc < 7; ++kc) {
#pragma unroll
        for (int j = 0; j < 2; ++j) {
            const int klo = kc * 4 + j;        // k when hi == 0
            const int khi = klo + 2;           // k when hi == 1
            const int off_lo = (klo / 5) * PADW + (klo % 5);
            const int off_hi = (khi / 5) * PADW + (khi % 5);
            b[kc][j] = bbase[hi ? off_hi : off_lo];
        }
    }

    v8f acc[5];
#pragma unroll
    for (int m = 0; m < 5; ++m) acc[m] = (v8f){};

#pragma unroll
    for (int kc = 0; kc < 7; ++kc) {
#pragma unroll
        for (int m = 0; m < 5; ++m) {
            v2f a = *(const v2f*)(abase + m * (16 * 28) + kc * 4);
            // 8 args: (neg_a, A, neg_b, B, c_mod, C, reuse_a, reuse_b)
            acc[m] = __builtin_amdgcn_wmma_f32_16x16x4_f32(
                false, a, false, b[kc], (short)0, acc[m], false, false);
        }
    }

    // theta-max: accumulator tile m in this lane holds group g = 2m + hi
#pragma unroll
    for (int m = 0; m < 5; ++m) {
        int g = 2 * m + hi;
        if (g < 9) {
            float mx = acc[m][0];
#pragma unroll
            for (int v = 1; v < 8; ++v) mx = fmaxf(mx, acc[m][v]);
            out9[(size_t)g * HW + (size_t)y * HH + x] = mx;
        }
    }
}

// ---------------------------------------------------------------------------
// Stage 3: per-channel mean over 496x496
// ---------------------------------------------------------------------------
__global__ void mean_kernel(const float* __restrict__ out9,
                            float* __restrict__ means) {
    __shared__ float s[256];
    int g = blockIdx.x;
    const float* ch = out9 + (size_t)g * HW;
    float sum = 0.0f;
    for (int i = threadIdx.x; i < HW; i += 256) sum += ch[i];
    s[threadIdx.x] = sum;
    __syncthreads();
    for (int off = 128; off > 0; off >>= 1) {
        if (threadIdx.x < off) s[threadIdx.x] += s[threadIdx.x + off];
        __syncthreads();
    }
    if (threadIdx.x == 0) means[g] = s[0] / (float)HW;
}

// ---------------------------------------------------------------------------
// Stage 4: per 16x16 window: first-argmax(thr) == 128 for any group -> valid
// ---------------------------------------------------------------------------
__global__ void winmax_kernel(const float* __restrict__ out9,
                              const float* __restrict__ means,
                              int* __restrict__ valid) {
    __shared__ float sv[256];
    __shared__ int   si[256];
    int w = blockIdx.x, t = threadIdx.x;
    int y = (w / NW) * 16 + t / 16;
    int x = (w % NW) * 16 + t % 16;
    int v_any = 0;
    for (int g = 0; g < 9; ++g) {
        float val = out9[(size_t)g * HW + y * HH + x];
        float thr = means[g] * 5.0f;
        sv[t] = (val > thr) ? val : 0.0f;
        si[t] = t;
        __syncthreads();
        for (int off = 128; off > 0; off >>= 1) {
            if (t < off) {
                float v2 = sv[t + off];
                int   i2 = si[t + off];
                // first occurrence of max (jnp.argmax semantics)
                if (v2 > sv[t] || (v2 == sv[t] && i2 < si[t])) {
                    sv[t] = v2;
                    si[t] = i2;
                }
            }
            __syncthreads();
        }
        if (t == 0 && si[0] == 128) v_any = 1;
        __syncthreads();
    }
    if (t == 0) valid[w] = v_any;
}

// ---------------------------------------------------------------------------
// Stage 5: gather 32x32 patches (zero-padded), nonzero-reduce, mask, store
// ---------------------------------------------------------------------------
__global__ void patch_kernel(const float* __restrict__ out9,
                             const int* __restrict__ valid,
                             float* __restrict__ out) {
    __shared__ int nz;
    int b = blockIdx.x;
    int g = b / NWIN;
    int w = b % NWIN;
    int t = threadIdx.x;
    if (t == 0) nz = 0;
    __syncthreads();
    int row0 = (w / NW) * 16 + 8 - 16;   // real-coordinate patch origin
    int col0 = (w % NW) * 16 + 8 - 16;
    float vals[4];
    int any = 0;
#pragma unroll
    for (int j = 0; j < 4; ++j) {
        int e = t + j * 256;
        int r = row0 + (e >> 5);
        int c = col0 + (e & 31);
        float v = 0.0f;
        if (r >= 0 && r < HH && c >= 0 && c < HH)
            v = out9[(size_t)g * HW + r * HH + c];
        vals[j] = v;
        any |= (v != 0.0f);
    }
    if (any) nz = 1;                      // benign race, same value
    __syncthreads();
    float m = (valid[w] && nz) ? 1.0f : 0.0f;
    size_t base = (size_t)b * 1024;
#pragma unroll
    for (int j = 0; j < 4; ++j) out[base + t + j * 256] = vals[j] * m;
}

// ---------------------------------------------------------------------------
extern "C" void kernel_launch(void* const* d_in, const int* in_sizes, int n_in,
                              void* d_out, int out_size, void* d_ws, size_t ws_size,
                              hipStream_t stream) {
    const float* img     = (const float*)d_in[0];   // 1x1x496x496
    const float* sigmas  = (const float*)d_in[1];   // 3
    const float* lambdas = (const float*)d_in[2];   // 3
    float* outp = (float*)d_out;                    // 9*961*32*32

    float* ws    = (float*)d_ws;
    float* Fp    = ws;                          // 80*28 = 2240 floats (pad 2304)
    float* padi  = ws + 2304;                   // 504*512 = 258048 floats
    float* out9  = padi + PADH * PADW;          // 9 * 246016 floats
    float* means = out9 + 9 * HW;               // 9 floats (pad 16)
    int*   valid = (int*)(means + 16);          // 961 ints

    gabor_kernel<<<1, 256, 0, stream>>>(sigmas, lambdas, Fp);
    pad_kernel<<<(PADH * PADW) / 256, 256, 0, stream>>>(img, padi);
    conv_wmma_kernel<<<1922, 256, 0, stream>>>(padi, Fp, out9);  // 15376 waves
    mean_kernel<<<9, 256, 0, stream>>>(out9, means);
    winmax_kernel<<<NWIN, 256, 0, stream>>>(out9, means, valid);
    patch_kernel<<<9 * NWIN, 256, 0, stream>>>(out9, valid, outp);
}